// Patch_loss_68307159875734
// MI455X (gfx1250) — compile-verified
//
#include <hip/hip_runtime.h>
#include <hip/hip_bf16.h>

typedef float v2f __attribute__((ext_vector_type(2)));
typedef float v8f __attribute__((ext_vector_type(8)));

// Problem constants (reference shapes are fixed)
constexpr int   Wd        = 512;          // image width
constexpr int   IMG       = 512 * 512;    // pixels per (b,c) image
constexpr int   NPIX      = 16 * 3 * IMG; // 12,582,912 total pixels
constexpr int   NPATCH    = 16 * 3 * 32 * 32; // 49,152 patches
constexpr float EPS       = 1e-6f;

// Reduce 8 f32 values/lane (256 per wave) to a wave-wide total using four
// chained V_WMMA_F32_16X16X4_F32 ops with an all-ones B matrix.
// A (16x4 f32) = 2 VGPRs; B(ones) makes every D column the per-row sum, so
// summing the 8 D VGPRs in-lane + one cross-half shuffle yields the total.
__device__ __forceinline__ float wave_patch_sum(float4 a, float4 b, v2f ones) {
    v8f acc = {};
    acc = __builtin_amdgcn_wmma_f32_16x16x4_f32(false, (v2f){a.x, a.y}, false, ones,
                                                (short)0, acc, false, false);
    acc = __builtin_amdgcn_wmma_f32_16x16x4_f32(false, (v2f){a.z, a.w}, false, ones,
                                                (short)0, acc, false, false);
    acc = __builtin_amdgcn_wmma_f32_16x16x4_f32(false, (v2f){b.x, b.y}, false, ones,
                                                (short)0, acc, false, false);
    acc = __builtin_amdgcn_wmma_f32_16x16x4_f32(false, (v2f){b.z, b.w}, false, ones,
                                                (short)0, acc, false, false);
    float s = acc[0] + acc[1] + acc[2] + acc[3] + acc[4] + acc[5] + acc[6] + acc[7];
    s += __shfl_xor(s, 16, 32);   // combine rows 0-7 (lanes<16) with rows 8-15
    return s;
}

__device__ __forceinline__ float4 f4mul(float4 a, float4 b) {
    return make_float4(a.x * b.x, a.y * b.y, a.z * b.z, a.w * b.w);
}

__global__ __launch_bounds__(256) void patch_loss_main(
        const float* __restrict__ x0,   // "out" tensor
        const float* __restrict__ x1,
        const float* __restrict__ x2,
        float* __restrict__ dout,       // [0]=loss, [1..]=w1_img, then w2_img
        float* __restrict__ ws)         // per-patch SSE partials
{
    const int lane = threadIdx.x & 31;
    const int wave = threadIdx.x >> 5;
    const int P    = blockIdx.x * 8 + wave;      // patch id, grid sized exactly

    // patch coordinates
    const int bc  = P >> 10;          // (b*3+c)
    const int rem = P & 1023;
    const int pi  = rem >> 5;         // patch row
    const int pj  = rem & 31;         // patch col
    const long base = (long)bc * IMG + (long)(pi * 16) * Wd + (long)(pj * 16);

    // each lane loads two float4 chunks (16 B each) per tensor: q = lane, lane+32
    const int q0 = lane, q1 = lane + 32;
    const long off0 = base + (long)(q0 >> 2) * Wd + (long)((q0 & 3) * 4);
    const long off1 = base + (long)(q1 >> 2) * Wd + (long)((q1 & 3) * 4);

    const float4 a0 = *(const float4*)(x0 + off0);
    const float4 b0 = *(const float4*)(x0 + off1);
    const float4 a1 = *(const float4*)(x1 + off0);
    const float4 b1 = *(const float4*)(x1 + off1);
    const float4 a2 = *(const float4*)(x2 + off0);
    const float4 b2 = *(const float4*)(x2 + off1);

    const v2f ones = {1.0f, 1.0f};

    // eight per-patch moment sums via WMMA reductions (full f32 precision)
    const float S1  = wave_patch_sum(a1, b1, ones);
    const float S2  = wave_patch_sum(a2, b2, ones);
    const float S11 = wave_patch_sum(f4mul(a1, a1), f4mul(b1, b1), ones);
    const float S22 = wave_patch_sum(f4mul(a2, a2), f4mul(b2, b2), ones);
    const float S00 = wave_patch_sum(f4mul(a0, a0), f4mul(b0, b0), ones);
    const float S01 = wave_patch_sum(f4mul(a0, a1), f4mul(b0, b1), ones);
    const float S02 = wave_patch_sum(f4mul(a0, a2), f4mul(b0, b2), ones);
    const float S12 = wave_patch_sum(f4mul(a1, a2), f4mul(b1, b2), ones);

    const float inv_p = 1.0f / 256.0f;
    const float mu1  = S1 * inv_p;
    const float var1 = fmaxf(S11 * inv_p - mu1 * mu1, 0.0f);
    const float sd1  = sqrtf(var1);
    // NOTE: reference (faithfully) centers patch2 around mu1 as well
    const float var2 = fmaxf(S22 * inv_p - 2.0f * mu1 * (S2 * inv_p) + mu1 * mu1, 0.0f);
    const float sd2  = sqrtf(var2);

    const float denom = sd1 + sd2 + EPS;
    const float w1 = sd1 / denom;
    const float w2 = sd2 / denom;

    // patch sum of squared error via algebraic expansion (single pass)
    const float sse = S00 + w1 * w1 * S11 + w2 * w2 * S22
                    - 2.0f * w1 * S01 - 2.0f * w2 * S02
                    + 2.0f * w1 * w2 * S12;

    // broadcast weights into the two weight images (coalesced scalar stores;
    // dout+1 is only 4B-aligned, so b32 stores striped across lanes)
    float* __restrict__ w1img = dout + 1;
    float* __restrict__ w2img = dout + 1 + NPIX;
#pragma unroll
    for (int s = 0; s < 8; ++s) {
        const int  p   = s * 32 + lane;               // pixel 0..255 in patch
        const long idx = base + (long)(p >> 4) * Wd + (long)(p & 15);
        w1img[idx] = w1;
        w2img[idx] = w2;
    }

    if (lane == 0) ws[P] = sse;
}

// Deterministic fixed-order final reduction of the 49,152 per-patch partials.
__global__ __launch_bounds__(256) void patch_loss_reduce(
        const float* __restrict__ ws, float* __restrict__ dout)
{
    __shared__ float red[256];
    const int t = threadIdx.x;
    float s = 0.0f;
    for (int k = t; k < NPATCH; k += 256) s += ws[k];   // fixed order per thread
    red[t] = s;
    __syncthreads();
#pragma unroll
    for (int off = 128; off > 0; off >>= 1) {
        if (t < off) red[t] += red[t + off];
        __syncthreads();
    }
    if (t == 0) dout[0] = red[0] * (1.0f / (float)NPIX);
}

extern "C" void kernel_launch(void* const* d_in, const int* in_sizes, int n_in,
                              void* d_out, int out_size, void* d_ws, size_t ws_size,
                              hipStream_t stream) {
    const float* x0 = (const float*)d_in[0];   // "out"
    const float* x1 = (const float*)d_in[1];
    const float* x2 = (const float*)d_in[2];
    float* out = (float*)d_out;
    float* ws  = (float*)d_ws;                 // needs NPATCH*4 = 192 KB

    // 8 waves per block, one wave per patch -> exactly NPATCH waves
    patch_loss_main<<<NPATCH / 8, 256, 0, stream>>>(x0, x1, x2, out, ws);
    patch_loss_reduce<<<1, 256, 0, stream>>>(ws, out);
}